// DictionaryLearningkSVD_76914274337307
// MI455X (gfx1250) — compile-verified
//
#include <hip/hip_runtime.h>
#include <math.h>

// ---------------- problem constants ----------------
#define NSIG   65536          // 64 * 32 * 32 signals
#define DIMN   64             // signal dim
#define KATOMS 512            // dictionary atoms
#define LNZ    8              // nonzeros per signal (OMP)
#define BETAC  0.25f
#define EPSV   1e-10f

// ---------------- workspace layout (float offsets) ----------------
#define WS_XN    ((size_t)0)            // 65536*64   = 4194304
#define WS_G     ((size_t)4194304)      // 512*512    = 262144
#define WS_ZDL   ((size_t)4456448)      // 65536*64   = 4194304
#define WS_PROB  ((size_t)8650752)      // 512
#define WS_LOSSP ((size_t)8651264)      // 8192 block partials
// total ~8.66M floats (~35 MB)

// ---------------- output layout (float offsets) ----------------
#define OUT_LOSS ((size_t)0)
#define OUT_ZST  ((size_t)1)
#define OUT_PERP ((size_t)4194305)
#define OUT_REP  ((size_t)4194306)      // 65536*512 floats; doubles as Alpha0 scratch

typedef float v2f __attribute__((ext_vector_type(2)));
typedef float v8f __attribute__((ext_vector_type(8)));

// =====================================================================
// Kernel 1: transpose (b,c,h,w)->(signal,c) and L2-normalize each row.
// =====================================================================
__global__ __launch_bounds__(256) void k_normalize(const float* __restrict__ ze,
                                                   float* __restrict__ xn)
{
    const int i  = blockIdx.x * 256 + threadIdx.x;       // signal id [0,65536)
    const int b  = i >> 10;                              // 1024 signals per batch
    const int hw = i & 1023;
    const float* src = ze + (size_t)b * (64u * 1024u) + hw;

    float v[DIMN];
    float ss = 0.0f;
#pragma unroll
    for (int c = 0; c < DIMN; ++c) {
        float t = src[(size_t)c * 1024];
        v[c] = t;
        ss += t * t;
    }
    const float inv = 1.0f / sqrtf(ss);
    float4* dst = (float4*)(xn + (size_t)i * DIMN);
#pragma unroll
    for (int q = 0; q < DIMN / 4; ++q)
        dst[q] = make_float4(v[4*q]*inv, v[4*q+1]*inv, v[4*q+2]*inv, v[4*q+3]*inv);
}

// =====================================================================
// fp32 WMMA 16x16 tile:  C[m0:+16, n0:+16] = Arows x Brows^T  (K = 64)
// A layout (16x4 fp32): lanes 0-15 hold M=lane, v0=K0,v1=K1; lanes 16-31 K2,K3
// B layout (4x16):      v0: half0->K0, half1->K2; v1: K1/K3; N = lane&15
// C/D (16x16 f32):      vgpr v: lanes0-15 M=v, lanes16-31 M=v+8; N = lane&15
// =====================================================================
__device__ __forceinline__ void wmma_tile_f32(const float* __restrict__ A, int lda,
                                              const float* __restrict__ B, int ldb,
                                              float* __restrict__ C, int ldc,
                                              int m0, int n0, int lane)
{
    const int half = lane >> 4;
    const int lq   = lane & 15;
    v8f acc;
#pragma unroll
    for (int q = 0; q < 8; ++q) acc[q] = 0.0f;

    const float* ap = A + (size_t)(m0 + lq) * lda + 2 * half;
    const float* bp = B + (size_t)(n0 + lq) * ldb + 2 * half;
#pragma unroll
    for (int k = 0; k < DIMN; k += 4) {
        v2f a = *(const v2f*)(ap + k);
        v2f b = *(const v2f*)(bp + k);
        acc = __builtin_amdgcn_wmma_f32_16x16x4_f32(false, a, false, b,
                                                    (short)0, acc, false, false);
    }
#pragma unroll
    for (int v = 0; v < 8; ++v)
        C[(size_t)(m0 + v + 8 * half) * ldc + (n0 + lq)] = acc[v];
}

// G = D * D^T   (512x512), 1 tile per wave, 32x32 tiles
__global__ __launch_bounds__(256) void k_gram(const float* __restrict__ Dict,
                                              float* __restrict__ G)
{
    const int wave = blockIdx.x * 8 + (threadIdx.x >> 5);
    const int lane = threadIdx.x & 31;
    const int ti = wave >> 5, tj = wave & 31;
    wmma_tile_f32(Dict, DIMN, Dict, DIMN, G, KATOMS, ti * 16, tj * 16, lane);
}

// Alpha0 = Xn * D^T  (65536x512) -> written into rep region of d_out
__global__ __launch_bounds__(256) void k_alpha(const float* __restrict__ Xn,
                                               const float* __restrict__ Dict,
                                               float* __restrict__ Alpha)
{
    const int wave = blockIdx.x * 8 + (threadIdx.x >> 5);
    const int lane = threadIdx.x & 31;
    const int ti = wave >> 5, tj = wave & 31;   // 4096 x 32 tiles
    wmma_tile_f32(Xn, DIMN, Dict, DIMN, Alpha, KATOMS, ti * 16, tj * 16, lane);
}

// =====================================================================
// Kernel 4: OMP, one wave32 per signal. Fully unrolled 8 iterations so the
// growing Cholesky solves have compile-time indices (pure VGPR, no scratch).
// Reads its Alpha0 row from `rep`, overwrites it with the sparse code.
// =====================================================================
#define LI(i, k) ((i) * ((i) + 1) / 2 + (k))

__global__ __launch_bounds__(256) void k_omp(const float* __restrict__ G,
                                             const float* __restrict__ Dict,
                                             const float* __restrict__ Xn,
                                             float* __restrict__ rep,
                                             float* __restrict__ zdl,
                                             float* __restrict__ lossPart)
{
    const int lane = threadIdx.x & 31;
    const int wib  = threadIdx.x >> 5;
    const int sig  = blockIdx.x * 8 + wib;

    // ---- load a0 (this signal's correlation row), 16 values per lane ----
    float a0[16], rv[16];
    {
        const float4* ap = (const float4*)(rep + (size_t)sig * KATOMS + lane * 16);
#pragma unroll
        for (int q = 0; q < 4; ++q) {
            float4 t = ap[q];
            a0[4*q] = t.x; a0[4*q+1] = t.y; a0[4*q+2] = t.z; a0[4*q+3] = t.w;
        }
    }

    int   idxA[LNZ];
    float coefA[LNZ], bv[LNZ];
    float Lm[36], drec[LNZ], y[LNZ];
#pragma unroll
    for (int s = 0; s < LNZ; ++s) { idxA[s] = 0; coefA[s] = 0.0f; bv[s] = 0.0f; }

#pragma unroll
    for (int t = 0; t < LNZ; ++t) {
        // resid = a0 - sum_{s<t} coef[s] * G[idx[s], :]   (G symmetric -> row reads)
#pragma unroll
        for (int q = 0; q < 16; ++q) rv[q] = a0[q];
#pragma unroll
        for (int s = 0; s < t; ++s) {
            const float c = coefA[s];
            const float4* gp = (const float4*)(G + (size_t)idxA[s] * KATOMS + lane * 16);
#pragma unroll
            for (int q = 0; q < 4; ++q) {
                float4 g = gp[q];
                rv[4*q]   -= c * g.x; rv[4*q+1] -= c * g.y;
                rv[4*q+2] -= c * g.z; rv[4*q+3] -= c * g.w;
            }
        }
        // local argmax of |resid| over this lane's 16 candidates (mask chosen)
        float bvv = -__builtin_inff();
        int   bi  = lane * 16;
        float ba  = 0.0f;
#pragma unroll
        for (int q = 0; q < 16; ++q) {
            const int gidx = lane * 16 + q;
            bool excl = false;
#pragma unroll
            for (int s = 0; s < t; ++s) excl = excl || (idxA[s] == gidx);
            const float vq = excl ? -__builtin_inff() : fabsf(rv[q]);
            if (vq > bvv) { bvv = vq; bi = gidx; ba = a0[q]; }   // ties -> lowest idx
        }
        // wave32 butterfly argmax (val desc, index asc on tie); also carries a0[j]
#pragma unroll
        for (int off = 16; off > 0; off >>= 1) {
            const float ov = __shfl_xor(bvv, off);
            const int   oi = __shfl_xor(bi, off);
            const float oa = __shfl_xor(ba, off);
            if (ov > bvv || (ov == bvv && oi < bi)) { bvv = ov; bi = oi; ba = oa; }
        }
        idxA[t] = bi;
        bv[t]   = ba;

        // Cholesky of Gsub[(t+1)x(t+1)]; entries are broadcast loads from G
#pragma unroll
        for (int i = 0; i <= t; ++i) {
#pragma unroll
            for (int k2 = 0; k2 < i; ++k2) {
                float s = G[(size_t)idxA[i] * KATOMS + idxA[k2]];
#pragma unroll
                for (int m = 0; m < k2; ++m) s -= Lm[LI(i, m)] * Lm[LI(k2, m)];
                Lm[LI(i, k2)] = s * drec[k2];
            }
            float s = G[(size_t)idxA[i] * KATOMS + idxA[i]];
#pragma unroll
            for (int m = 0; m < i; ++m) s -= Lm[LI(i, m)] * Lm[LI(i, m)];
            const float d = sqrtf(fmaxf(s, 1e-20f));
            Lm[LI(i, i)] = d;
            drec[i] = 1.0f / d;
        }
        // forward solve L y = b
#pragma unroll
        for (int i = 0; i <= t; ++i) {
            float s = bv[i];
#pragma unroll
            for (int m = 0; m < i; ++m) s -= Lm[LI(i, m)] * y[m];
            y[i] = s * drec[i];
        }
        // backward solve L^T coef = y
#pragma unroll
        for (int ii = t; ii >= 0; --ii) {
            float s = y[ii];
#pragma unroll
            for (int m = ii + 1; m <= t; ++m) s -= Lm[LI(m, ii)] * coefA[m];
            coefA[ii] = s * drec[ii];
        }
    }

    // ---- scatter sparse code back over our a0 row (rep output) ----
    {
        float4* op = (float4*)(rep + (size_t)sig * KATOMS + lane * 16);
#pragma unroll
        for (int q = 0; q < 4; ++q) {
            float4 w = make_float4(0.f, 0.f, 0.f, 0.f);
            const int base = lane * 16 + 4 * q;
#pragma unroll
            for (int s = 0; s < LNZ; ++s) {
                const float c = coefA[s];
                const int id = idxA[s];
                w.x = (id == base + 0) ? c : w.x;
                w.y = (id == base + 1) ? c : w.y;
                w.z = (id == base + 2) ? c : w.z;
                w.w = (id == base + 3) ? c : w.w;
            }
            op[q] = w;
        }
    }

    // ---- z_dl = sum coef[s] * Dict[idx[s]]  (2 channels per lane) ----
    float z0 = 0.f, z1 = 0.f;
#pragma unroll
    for (int s = 0; s < LNZ; ++s) {
        const float2 dv = *(const float2*)(Dict + (size_t)idxA[s] * DIMN + 2 * lane);
        z0 += coefA[s] * dv.x;
        z1 += coefA[s] * dv.y;
    }
    *(float2*)(zdl + (size_t)sig * DIMN + 2 * lane) = make_float2(z0, z1);

    // ---- per-block deterministic loss partial: sum (z_dl - z)^2 ----
    const float2 xv = *(const float2*)(Xn + (size_t)sig * DIMN + 2 * lane);
    const float d0 = z0 - xv.x, d1 = z1 - xv.y;
    float ls = d0 * d0 + d1 * d1;
#pragma unroll
    for (int off = 16; off > 0; off >>= 1) ls += __shfl_xor(ls, off);

    __shared__ float shl[8];
    if (lane == 0) shl[wib] = ls;
    __syncthreads();
    if (threadIdx.x == 0) {
        float tot = 0.f;
#pragma unroll
        for (int w = 0; w < 8; ++w) tot += shl[w];
        lossPart[blockIdx.x] = tot;
    }
}

// =====================================================================
// Kernel 5: deterministic column sums of rep -> probs[512]
// =====================================================================
__global__ __launch_bounds__(256) void k_probs(const float* __restrict__ rep,
                                               float* __restrict__ probs)
{
    const int k = blockIdx.x;  // atom
    float s = 0.f;
    for (int r = threadIdx.x; r < NSIG; r += 256)
        s += rep[(size_t)r * KATOMS + k];
#pragma unroll
    for (int off = 16; off > 0; off >>= 1) s += __shfl_xor(s, off);
    __shared__ float sh[8];
    const int lane = threadIdx.x & 31, wib = threadIdx.x >> 5;
    if (lane == 0) sh[wib] = s;
    __syncthreads();
    if (threadIdx.x == 0) {
        float t = 0.f;
#pragma unroll
        for (int w = 0; w < 8; ++w) t += sh[w];
        probs[k] = t;
    }
}

// =====================================================================
// Kernel 6: z_st = z_dl transposed back to (b, c, h, w); coalesced writes
// =====================================================================
__global__ __launch_bounds__(256) void k_zst(const float* __restrict__ zdl,
                                             float* __restrict__ out)
{
    const int o   = blockIdx.x * 256 + threadIdx.x;   // [0, 4194304)
    const int b   = o >> 16;
    const int rem = o & 65535;
    const int c   = rem >> 10;
    const int hw  = rem & 1023;
    out[o] = zdl[((size_t)((b << 10) | hw)) * DIMN + c];
}

// =====================================================================
// Kernel 7: finalize scalars (deterministic fixed-order reductions)
// =====================================================================
__device__ __forceinline__ float waveAdd(float v)
{
#pragma unroll
    for (int off = 16; off > 0; off >>= 1) v += __shfl_xor(v, off);
    return v;
}

__global__ __launch_bounds__(512) void k_final(const float* __restrict__ probs,
                                               const float* __restrict__ lossPart,
                                               float* __restrict__ out)
{
    __shared__ float sh[16];
    __shared__ float shb;
    const int t = threadIdx.x, lane = t & 31, wib = t >> 5;

    // total of atom sums
    const float s = probs[t];
    float r = waveAdd(s);
    if (lane == 0) sh[wib] = r;
    __syncthreads();
    if (t == 0) {
        float acc = 0.f;
#pragma unroll
        for (int w = 0; w < 16; ++w) acc += sh[w];
        shb = acc;
    }
    __syncthreads();
    const float total = shb;
    __syncthreads();

    // entropy term
    const float p = s / total;
    float e = p * logf(p + EPSV);
    e = waveAdd(e);
    if (lane == 0) sh[wib] = e;
    __syncthreads();
    float ent = 0.f;
    if (t == 0) {
#pragma unroll
        for (int w = 0; w < 16; ++w) ent += sh[w];
    }
    __syncthreads();

    // loss total
    float lp = 0.f;
    for (int i = t; i < 8192; i += 512) lp += lossPart[i];
    lp = waveAdd(lp);
    if (lane == 0) sh[wib] = lp;
    __syncthreads();
    if (t == 0) {
        float lt = 0.f;
#pragma unroll
        for (int w = 0; w < 16; ++w) lt += sh[w];
        out[OUT_LOSS] = (1.0f + BETAC) * lt / 4194304.0f;  // (1+beta)*MSE
        out[OUT_PERP] = expf(-ent);
    }
}

// =====================================================================
extern "C" void kernel_launch(void* const* d_in, const int* in_sizes, int n_in,
                              void* d_out, int out_size, void* d_ws, size_t ws_size,
                              hipStream_t stream)
{
    (void)in_sizes; (void)n_in; (void)out_size; (void)ws_size;

    const float* ze   = (const float*)d_in[0];   // (64, 64, 32, 32)
    const float* dict = (const float*)d_in[1];   // (512, 64)
    float* out = (float*)d_out;
    float* ws  = (float*)d_ws;

    float* Xn    = ws + WS_XN;
    float* G     = ws + WS_G;
    float* zdl   = ws + WS_ZDL;
    float* probs = ws + WS_PROB;
    float* lossP = ws + WS_LOSSP;
    float* rep   = out + OUT_REP;                // Alpha0 scratch, then rep output

    k_normalize<<<dim3(256),   dim3(256), 0, stream>>>(ze, Xn);
    k_gram     <<<dim3(128),   dim3(256), 0, stream>>>(dict, G);
    k_alpha    <<<dim3(16384), dim3(256), 0, stream>>>(Xn, dict, rep);
    k_omp      <<<dim3(8192),  dim3(256), 0, stream>>>(G, dict, Xn, rep, zdl, lossP);
    k_probs    <<<dim3(512),   dim3(256), 0, stream>>>(rep, probs);
    k_zst      <<<dim3(16384), dim3(256), 0, stream>>>(zdl, out + OUT_ZST);
    k_final    <<<dim3(1),     dim3(512), 0, stream>>>(probs, lossP, out);
}